// XBlock_27608049779425
// MI455X (gfx1250) — compile-verified
//
#include <hip/hip_runtime.h>

// Problem constants
#define BATCH 8
#define C 128
#define NROWS 36864           // H*W = 192*192
#define NF 36864.0f
#define INV_SQRT_DK 0.08838834764831845f   // 1/sqrt(128)

typedef __attribute__((ext_vector_type(2))) float v2f;
typedef __attribute__((ext_vector_type(8))) float v8f;

// ---------------------------------------------------------------------------
// Kernel 1: Gram matrix G[b] = x^T x (128x128) and column sums S[b] = x^T 1.
// x = images[b] viewed as [N,128] row-major. Double-buffered LDS tiles filled
// with GLOBAL_LOAD_ASYNC_TO_LDS_B32 (transpose scatter via per-lane LDS dest
// addresses, ASYNCcnt-tracked) so tile i+1 streams in while tile i feeds the
// f32 WMMAs. Wave w owns G rows [16w,16w+16) x all 128 cols (8 accum tiles)
// plus a 9th ones-column WMMA that yields the channel sums S.
// ---------------------------------------------------------------------------
#define GRAM_CHUNKS 32
#define TILE_ROWS 64
#define XT_STRIDE 72                        // 64 + 8 pad (floats)
#define TILE_LDS_FLOATS (C * XT_STRIDE)     // 9216 floats = 36864 B per buffer

__global__ __launch_bounds__(256) void gram_kernel(const float* __restrict__ images,
                                                   float* __restrict__ G,
                                                   float* __restrict__ S) {
  extern __shared__ float smem[];           // 2 * TILE_LDS_FLOATS (73,728 B)

  const int b    = blockIdx.y;
  const int t    = threadIdx.x;
  const int lane = t & 31;
  const int wv   = t >> 5;
  const float* xb = images + (size_t)b * NROWS * C;

  const int rows_per_chunk = NROWS / GRAM_CHUNKS;   // 1152
  const int row_base = blockIdx.x * rows_per_chunk;
  const int ntiles = rows_per_chunk / TILE_ROWS;    // 18

  const int c0  = wv * 16;
  const int li  = lane & 15;
  const int kkb = (lane >> 4) * 2;                  // K sub-pair (f32 A/B layout)

  // Issue one 64x128 tile: wave w covers rows w*8..w*8+7, 4 col-groups of 32.
  // Each async b32 reads 128 contiguous global bytes per wave and scatters
  // into the transposed LDS tile xtT[col][row].
  auto issue_tile = [&](int r0, int bufIdx) {
    const float* gbase = xb + (size_t)r0 * C;
    float* buf = &smem[bufIdx * TILE_LDS_FLOATS];
    #pragma unroll
    for (int j = 0; j < 32; ++j) {
      const int row = wv * 8 + (j >> 2);
      const int col = (j & 3) * 32 + lane;
      const uint32_t lds_off = (uint32_t)(uintptr_t)&buf[col * XT_STRIDE + row];
      const uint32_t goff    = (uint32_t)(row * C + col) * 4u;
      asm volatile("global_load_async_to_lds_b32 %0, %1, %2"
                   :: "v"(lds_off), "v"(goff), "s"(gbase) : "memory");
    }
  };

  v8f acc[8] = {};
  v8f accs   = {};
  const v2f ones = {1.0f, 1.0f};

  issue_tile(row_base, 0);
  for (int tile = 0; tile < ntiles; ++tile) {
    asm volatile("s_wait_asynccnt 0x0" ::: "memory");   // this wave's tile landed
    __syncthreads();                                    // all waves' portions landed
    if (tile + 1 < ntiles)                              // prefetch next tile (other buffer)
      issue_tile(row_base + (tile + 1) * TILE_ROWS, (tile + 1) & 1);

    const float* buf = &smem[(tile & 1) * TILE_LDS_FLOATS];
    for (int ks = 0; ks < TILE_ROWS / 4; ++ks) {
      const int k4 = ks * 4;
      // A[i][kk] = x[k4+kk][c0+i]
      const v2f a = *(const v2f*)&buf[(c0 + li) * XT_STRIDE + k4 + kkb];
      // ones-column: channel sums for rows c0..c0+15
      accs = __builtin_amdgcn_wmma_f32_16x16x4_f32(
          false, a, false, ones, (short)0, accs, false, false);
      #pragma unroll
      for (int db = 0; db < 8; ++db) {
        // B[kk][j] = x[k4+kk][16*db+j]
        const v2f bm = *(const v2f*)&buf[(db * 16 + li) * XT_STRIDE + k4 + kkb];
        acc[db] = __builtin_amdgcn_wmma_f32_16x16x4_f32(
            false, a, false, bm, (short)0, acc[db], false, false);
      }
    }
    __syncthreads();
  }

  // Accumulate partial Gram (C/D layout: vgpr r, lane L -> row r+8*(L>=16), col L&15)
  {
    float* Gb = G + (size_t)b * C * C;
    const int rofs = (lane >> 4) * 8;
    #pragma unroll
    for (int db = 0; db < 8; ++db)
      #pragma unroll
      for (int r = 0; r < 8; ++r)
        atomicAdd(&Gb[(size_t)(c0 + r + rofs) * C + db * 16 + li], acc[db][r]);
    if (li == 0) {
      #pragma unroll
      for (int r = 0; r < 8; ++r)
        atomicAdd(&S[b * C + c0 + r + rofs], accs[r]);
    }
  }
}

// ---------------------------------------------------------------------------
// Kernel 2: P[b] = [Wq|bq] @ G~[b]   (128 x 129), G~ = [[G, s],[s^T, N]]
// ---------------------------------------------------------------------------
__global__ void p_kernel(const float* __restrict__ Wq, const float* __restrict__ bq,
                         const float* __restrict__ G, const float* __restrict__ S,
                         float* __restrict__ P) {
  const int b = blockIdx.x;
  const float* Gb = G + (size_t)b * C * C;
  const float* Sb = S + b * C;
  float* Pb = P + (size_t)b * C * 129;
  for (int idx = threadIdx.x; idx < C * 129; idx += blockDim.x) {
    const int c = idx / 129, f = idx - c * 129;
    float acc = 0.f;
    const float* wr = Wq + c * C;
    if (f < C) {
      const float* g = Gb + f * C;          // G symmetric: G[e][f] = G[f][e]
      for (int e = 0; e < C; ++e) acc += wr[e] * g[e];
      acc += bq[c] * Sb[f];
    } else {
      for (int e = 0; e < C; ++e) acc += wr[e] * Sb[e];
      acc += bq[c] * NF;
    }
    Pb[idx] = acc;
  }
}

// ---------------------------------------------------------------------------
// Kernel 3: att[b] = softmax( (P @ [Wk|bk]^T) * 1/sqrt(DK), axis=-1 )
// ---------------------------------------------------------------------------
__global__ void att_kernel(const float* __restrict__ Wk, const float* __restrict__ bk,
                           const float* __restrict__ P, float* __restrict__ ATT) {
  const int b = blockIdx.x;
  const int c = threadIdx.x;                 // blockDim = 128
  const float* Pr = P + (size_t)b * C * 129 + c * 129;
  float* Ar = ATT + (size_t)b * C * C + c * C;
  float mx = -3.0e38f;
  for (int d = 0; d < C; ++d) {
    float acc = Pr[C] * bk[d];
    const float* wr = Wk + d * C;
    for (int f = 0; f < C; ++f) acc += Pr[f] * wr[f];
    acc *= INV_SQRT_DK;
    Ar[d] = acc;
    mx = fmaxf(mx, acc);
  }
  float denom = 0.f;
  for (int d = 0; d < C; ++d) { const float e = expf(Ar[d] - mx); Ar[d] = e; denom += e; }
  const float inv = 1.0f / denom;
  for (int d = 0; d < C; ++d) Ar[d] *= inv;
}

// ---------------------------------------------------------------------------
// Kernel 4: M~[b] = [Wv|bv]^T @ att[b]   (129 x 128); row 128 is m = att^T bv
// ---------------------------------------------------------------------------
__global__ void m_kernel(const float* __restrict__ Wv, const float* __restrict__ bv,
                         const float* __restrict__ ATT, float* __restrict__ MM) {
  const int b = blockIdx.x;
  const float* Ab = ATT + (size_t)b * C * C;
  float* Mb = MM + (size_t)b * 129 * C;
  for (int idx = threadIdx.x; idx < 129 * C; idx += blockDim.x) {
    const int e = idx >> 7, d = idx & 127;
    float acc = 0.f;
    if (e < C) {
      for (int c = 0; c < C; ++c) acc += Wv[c * C + e] * Ab[c * C + d];
    } else {
      for (int c = 0; c < C; ++c) acc += bv[c] * Ab[c * C + d];
    }
    Mb[idx] = acc;
  }
}

// ---------------------------------------------------------------------------
// Kernel 5: out = images + x @ M + 1 m^T. x tile staged via async b128
// (16B-aligned per-lane LDS dests); M~ transposed on the normal path.
// Residual add reuses the staged x tile (it *is* the image data).
// ---------------------------------------------------------------------------
#define AP_STRIDE 136                        // padded row stride in floats

__global__ __launch_bounds__(256) void apply_kernel(const float* __restrict__ images,
                                                    const float* __restrict__ MM,
                                                    float* __restrict__ out) {
  extern __shared__ float smem[];
  float* xt   = smem;                        // [128][AP_STRIDE] x tile, row-major
  float* Mt   = smem + C * AP_STRIDE;        // [128][AP_STRIDE] Mt[d][k] = M[k][d]
  float* mrow = Mt + C * AP_STRIDE;          // [128] bias row m

  const int b    = blockIdx.y;
  const int t    = threadIdx.x;
  const int lane = t & 31;
  const int wv   = t >> 5;
  const int row0 = blockIdx.x * 128;
  const float* xrow0 = images + (size_t)b * NROWS * C + (size_t)row0 * C;
  const float* Mb    = MM + (size_t)b * 129 * C;

  // stage 128x128 x tile via async b128 -> LDS
  #pragma unroll
  for (int i = 0; i < 16; ++i) {
    const int f4   = i * 256 + t;            // 0..4095 float4s
    const int row  = f4 >> 5;
    const int col4 = (f4 & 31) * 4;
    const uint32_t lds_off = (uint32_t)(uintptr_t)&xt[row * AP_STRIDE + col4];
    const uint32_t goff    = (uint32_t)(row * C + col4) * 4u;
    asm volatile("global_load_async_to_lds_b128 %0, %1, %2"
                 :: "v"(lds_off), "v"(goff), "s"(xrow0) : "memory");
  }
  // stage M~ (129x128) transposed; row 128 -> mrow
  for (int i = 0; i < 65; ++i) {
    const int idx = i * 256 + t;
    if (idx < 129 * C) {
      const int k = idx >> 7, d = idx & 127;
      const float v = Mb[idx];
      if (k < C) Mt[d * AP_STRIDE + k] = v;
      else       mrow[d] = v;
    }
  }
  asm volatile("s_wait_asynccnt 0x0" ::: "memory");
  __syncthreads();

  v8f acc[8] = {};
  const int li  = lane & 15;
  const int kkb = (lane >> 4) * 2;
  for (int ks = 0; ks < 32; ++ks) {
    const int k4 = ks * 4;
    const v2f a = *(const v2f*)&xt[(wv * 16 + li) * AP_STRIDE + k4 + kkb];
    #pragma unroll
    for (int db = 0; db < 8; ++db) {
      const v2f bm = *(const v2f*)&Mt[(db * 16 + li) * AP_STRIDE + k4 + kkb];
      acc[db] = __builtin_amdgcn_wmma_f32_16x16x4_f32(
          false, a, false, bm, (short)0, acc[db], false, false);
    }
  }

  float* ob = out + (size_t)b * NROWS * C + (size_t)row0 * C;
  const int rofs = (lane >> 4) * 8;
  #pragma unroll
  for (int db = 0; db < 8; ++db) {
    const int col = db * 16 + li;
    const float mv = mrow[col];
    #pragma unroll
    for (int r = 0; r < 8; ++r) {
      const int rowb = wv * 16 + r + rofs;
      ob[(size_t)rowb * C + col] = xt[rowb * AP_STRIDE + col] + acc[db][r] + mv;
    }
  }
}

// ---------------------------------------------------------------------------
extern "C" void kernel_launch(void* const* d_in, const int* in_sizes, int n_in,
                              void* d_out, int out_size, void* d_ws, size_t ws_size,
                              hipStream_t stream) {
  const float* images = (const float*)d_in[0];
  const float* Wq = (const float*)d_in[1];
  const float* bq = (const float*)d_in[2];
  const float* Wk = (const float*)d_in[3];
  const float* bk = (const float*)d_in[4];
  const float* Wv = (const float*)d_in[5];
  const float* bv = (const float*)d_in[6];
  float* out = (float*)d_out;

  // workspace layout (floats): ~2.1 MB total
  float* G   = (float*)d_ws;                 // B*128*128
  float* S   = G + BATCH * C * C;            // B*128
  float* P   = S + BATCH * C;                // B*128*129
  float* ATT = P + BATCH * C * 129;          // B*128*128
  float* MM  = ATT + BATCH * C * C;          // B*129*128

  hipMemsetAsync(G, 0, (size_t)(BATCH * C * C + BATCH * C) * sizeof(float), stream);

  const size_t gram_smem = (size_t)2 * TILE_LDS_FLOATS * sizeof(float);   // 73,728 B
  gram_kernel<<<dim3(GRAM_CHUNKS, BATCH), dim3(256), gram_smem, stream>>>(images, G, S);
  p_kernel<<<dim3(BATCH), dim3(256), 0, stream>>>(Wq, bq, G, S, P);
  att_kernel<<<dim3(BATCH), dim3(128), 0, stream>>>(Wk, bk, P, ATT);
  m_kernel<<<dim3(BATCH), dim3(256), 0, stream>>>(Wv, bv, ATT, MM);

  const size_t apply_smem = (size_t)(2 * C * AP_STRIDE + C) * sizeof(float); // 139,776 B
  apply_kernel<<<dim3(NROWS / 128, BATCH), dim3(256), apply_smem, stream>>>(images, MM, out);
}